// BloomEmbed_24318104830309
// MI455X (gfx1250) — compile-verified
//
#include <hip/hip_runtime.h>
#include <stdint.h>

// BloomEmbed: out[token, :] = (1/8) * sum_k W[mueller_hash(t[token]+k) % 1e6, :]
//
// Memory-bound random gather: ~302 MB of HBM traffic -> ~13 us floor at 23.3 TB/s.
// Strategy: one wave32 per token. Each lane owns a 16B (float4) slice of the
// 128-dim row, so each 512B-row gather is a single perfectly-coalesced
// global_load_async_to_lds_b128 (CDNA5 async-to-LDS path, ASYNCcnt-tracked)
// into a per-wave 4 KB LDS staging buffer. One s_wait_asynccnt 0, then 8x
// ds_load_b128 + f32 accumulate, scale by 1/8, store b128.
//
// WMMA is intentionally not used: the gathered rows are per-token (no shared
// matmul operand), and the validated output is f32, so an exact reduction is
// 7 VALU adds per float4 — cheaper than any 16x16x4_f32 WMMA formulation.

#define NUMROWS   1000000
#define DIM       128
#define KPROBE    8
#define WAVES_PB  8        // 256 threads / block
#define ROWBYTES  (DIM * 4)

__device__ __forceinline__ int64_t mueller_hash(int64_t t) {
    // Match torch/jax int64 semantics: arithmetic shift, XOR, wrapping multiply.
    t = (int64_t)((uint64_t)((t >> 16) ^ t) * 73244475ull);
    t = (int64_t)((uint64_t)((t >> 16) ^ t) * 73244475ull);
    t = (t >> 16) ^ t;
    return t;
}

__global__ __launch_bounds__(32 * WAVES_PB)
void bloom_embed_kernel(const int64_t* __restrict__ tok,
                        const float*   __restrict__ W,
                        float*         __restrict__ out,
                        int nTok) {
    __shared__ alignas(16) float buf[WAVES_PB][KPROBE][DIM];  // 32 KB / block

    const int lane  = threadIdx.x & 31;
    const int wv    = threadIdx.x >> 5;
    const int token = blockIdx.x * WAVES_PB + wv;
    if (token >= nTok) return;   // wave-uniform exit (token is per-wave)

    // Lanes 0..7 (replicated x4) each compute one of the 8 hash probes.
    const int64_t tv = tok[token];
    int64_t h = mueller_hash(tv + (int64_t)(lane & (KPROBE - 1)));
    int idx = (int)(h % (int64_t)NUMROWS);
    if (idx < 0) idx += NUMROWS;          // Python floor-mod (sign of divisor)
    int rowByte = idx << 9;               // idx * 512, fits in i32 (< 2^29)

    // Workgroup-relative LDS byte address of this wave's staging buffer:
    // low 32 bits of the flat shared pointer are the LDS offset.
    unsigned ldsBase = (unsigned)(size_t)(&buf[wv][0][0]);

    // Issue all 8 row gathers async to LDS (8 outstanding 512B transfers/wave).
#pragma unroll
    for (int k = 0; k < KPROBE; ++k) {
        int      gOff = __shfl(rowByte, k, 32) + lane * 16;          // byte off into W
        unsigned lOff = ldsBase + (unsigned)(k * ROWBYTES + lane * 16);
        asm volatile("global_load_async_to_lds_b128 %0, %1, %2"
                     :: "v"(lOff), "v"(gOff), "s"(W)
                     : "memory");
    }
    asm volatile("s_wait_asynccnt 0" ::: "memory");

    // Reduce the 8 staged rows (ds_load_b128) and average.
    float4 acc = make_float4(0.f, 0.f, 0.f, 0.f);
#pragma unroll
    for (int k = 0; k < KPROBE; ++k) {
        const float4 v = *(const float4*)&buf[wv][k][lane * 4];
        acc.x += v.x; acc.y += v.y; acc.z += v.z; acc.w += v.w;
    }
    acc.x *= 0.125f; acc.y *= 0.125f; acc.z *= 0.125f; acc.w *= 0.125f;

    *(float4*)&out[(size_t)token * DIM + lane * 4] = acc;
}

extern "C" void kernel_launch(void* const* d_in, const int* in_sizes, int n_in,
                              void* d_out, int out_size, void* d_ws, size_t ws_size,
                              hipStream_t stream) {
    // Inputs per reference: t int64 [32,2048], W float32 [1e6,128]. Output f32.
    const int64_t* tok = (const int64_t*)d_in[0];
    const float*   W   = (const float*)d_in[1];
    float*         out = (float*)d_out;

    const int nTok   = in_sizes[0];                      // 65536
    const int blocks = (nTok + WAVES_PB - 1) / WAVES_PB; // 8192

    bloom_embed_kernel<<<blocks, 32 * WAVES_PB, 0, stream>>>(tok, W, out, nTok);
}